// Attention_34772055228636
// MI455X (gfx1250) — compile-verified
//
#include <hip/hip_runtime.h>

typedef float v2f __attribute__((ext_vector_type(2)));
typedef float v8f __attribute__((ext_vector_type(8)));

#define BB 2
#define HH 16
#define SS 2048
#define DD 64
#define QTILE 16
#define SPAD 2056            // 2048 + 8 pad: keeps rows 8B aligned, shifts banks
#define NWAVE 4
#define CPW (SS / NWAVE)     // 512 key columns per wave

__global__ __launch_bounds__(128, 1)
void attn_f32_wmma(const float* __restrict__ Q, const float* __restrict__ K,
                   const float* __restrict__ V, const unsigned char* __restrict__ Mask,
                   float* __restrict__ out_ctx, float* __restrict__ out_attn) {
    extern __shared__ float smem[];
    float* sc      = smem;                   // [16][SPAD] scores -> probs
    float* ctxbuf  = sc + QTILE * SPAD;      // [16][64] context accumulator
    float* partred = ctxbuf + QTILE * DD;    // [16][8] per-row partials
    float* rowmax  = partred + 128;          // [16]
    float* rowinv  = rowmax + 16;            // [16]

    const int tid  = threadIdx.x;
    const int wave = tid >> 5;
    const int lane = tid & 31;
    const int half = lane >> 4;   // 0: lanes 0-15, 1: lanes 16-31
    const int lm   = lane & 15;

    const int qt = blockIdx.x & (SS / QTILE - 1);   // q-tile fastest -> L2 reuse of K/V
    const int bh = blockIdx.x >> 7;                 // S/QTILE == 128
    const int q0 = qt * QTILE;

    const float* Qb = Q + ((size_t)bh * SS + q0) * DD;
    const float* Kb = K + (size_t)bh * SS * DD;
    const float* Vb = V + (size_t)bh * SS * DD;
    const unsigned char* Mb = Mask + ((size_t)bh * SS + q0) * SS;
    float* ctx_out  = out_ctx  + ((size_t)bh * SS + q0) * DD;
    float* attn_out = out_attn + ((size_t)bh * SS + q0) * SS;

    // zero the context accumulator
    for (int i = tid; i < QTILE * DD; i += 128) ctxbuf[i] = 0.0f;

    // ---- Q tile (16x64) as 16 A-fragments of V_WMMA_F32_16X16X4_F32 ----
    // A 16x4 f32 layout: M = lane%16; VGPR0 holds K = 2*half, VGPR1 holds K = 2*half+1
    v2f aq[16];
#pragma unroll
    for (int i = 0; i < 16; ++i)
        aq[i] = *(const v2f*)(Qb + (size_t)lm * DD + 4 * i + 2 * half);

    // ---- scores = Q K^T * 1/sqrt(64), masked, into LDS ----
    const int cbase = wave * CPW;
#pragma unroll 1
    for (int t = 0; t < CPW / 16; ++t) {
        const int c0 = cbase + t * 16;
        v8f acc = {0.f, 0.f, 0.f, 0.f, 0.f, 0.f, 0.f, 0.f};
        // B 4x16 f32 layout: N = lane%16; VGPR0 holds K = 2*half, VGPR1 K = 2*half+1
        const float* kp = Kb + (size_t)(c0 + lm) * DD + 2 * half;
#pragma unroll
        for (int i = 0; i < 16; ++i) {
            v2f bk = *(const v2f*)(kp + 4 * i);
            acc = __builtin_amdgcn_wmma_f32_16x16x4_f32(
                false, aq[i], false, bk, (short)0, acc, false, false);
        }
        // C layout: value v -> row = v + 8*half, col = c0 + lane%16
#pragma unroll
        for (int v = 0; v < 8; ++v) {
            const int row = v + 8 * half;
            const int col = c0 + lm;
            float s = acc[v] * 0.125f;                 // 1/sqrt(64)
            if (Mb[(size_t)row * SS + col]) s = -1.0e9f;
            sc[row * SPAD + col] = s;
        }
    }
    __syncthreads();

    // ---- softmax: row max ----
    {
        const int row = tid >> 3, sub = tid & 7;
        float m = -3.0e38f;
        for (int c = sub * 256; c < sub * 256 + 256; ++c)
            m = fmaxf(m, sc[row * SPAD + c]);
        partred[row * 8 + sub] = m;
    }
    __syncthreads();
    if (tid < QTILE) {
        float m = partred[tid * 8];
#pragma unroll
        for (int j = 1; j < 8; ++j) m = fmaxf(m, partred[tid * 8 + j]);
        rowmax[tid] = m;
    }
    __syncthreads();
    // ---- softmax: exp + row sum ----
    {
        const int row = tid >> 3, sub = tid & 7;
        const float rm = rowmax[row];
        float s = 0.0f;
        for (int c = sub * 256; c < sub * 256 + 256; ++c) {
            float p = __expf(sc[row * SPAD + c] - rm);
            sc[row * SPAD + c] = p;
            s += p;
        }
        partred[row * 8 + sub] = s;
    }
    __syncthreads();
    if (tid < QTILE) {
        float s = 0.0f;
#pragma unroll
        for (int j = 0; j < 8; ++j) s += partred[tid * 8 + j];
        rowinv[tid] = 1.0f / s;
    }
    __syncthreads();

    // ---- context partial: P(16 x 512) @ V(512 x 64) per wave via f32 WMMA ----
    v8f cc[4];
#pragma unroll
    for (int d = 0; d < 4; ++d) cc[d] = (v8f){0.f, 0.f, 0.f, 0.f, 0.f, 0.f, 0.f, 0.f};
#pragma unroll 1
    for (int k = 0; k < CPW; k += 4) {
        const int kr = cbase + k + 2 * half;
        v2f ap = *(const v2f*)(&sc[lm * SPAD + kr]);        // A: P[m][kr], P[m][kr+1]
        const float* vp = Vb + (size_t)kr * DD + lm;
#pragma unroll
        for (int d = 0; d < 4; ++d) {
            v2f bv;
            bv.x = vp[d * 16];                               // V[kr  ][d*16+lm]
            bv.y = vp[DD + d * 16];                          // V[kr+1][d*16+lm]
            cc[d] = __builtin_amdgcn_wmma_f32_16x16x4_f32(
                false, ap, false, bv, (short)0, cc[d], false, false);
        }
    }
    // merge per-wave partials (ds_add_f32)
#pragma unroll
    for (int d = 0; d < 4; ++d)
#pragma unroll
        for (int v = 0; v < 8; ++v) {
            const int row = v + 8 * half;
            atomicAdd(&ctxbuf[row * DD + d * 16 + lm], cc[d][v]);
        }

    // ---- write normalized attn (coalesced 128-thread sweep) ----
    for (int idx = tid; idx < QTILE * SS; idx += 128) {
        const int r = idx >> 11;
        const int c = idx & (SS - 1);
        attn_out[(size_t)r * SS + c] = sc[r * SPAD + c] * rowinv[r];
    }
    __syncthreads();

    // ---- write normalized context ----
    for (int idx = tid; idx < QTILE * DD; idx += 128) {
        const int r = idx >> 6;
        ctx_out[idx] = ctxbuf[idx] * rowinv[r];
    }
}

extern "C" void kernel_launch(void* const* d_in, const int* in_sizes, int n_in,
                              void* d_out, int out_size, void* d_ws, size_t ws_size,
                              hipStream_t stream) {
    const float* Q = (const float*)d_in[0];
    const float* K = (const float*)d_in[1];
    const float* V = (const float*)d_in[2];
    const unsigned char* Mask = (const unsigned char*)d_in[3];  // jnp.bool_ -> 1 byte

    float* out_ctx  = (float*)d_out;                                  // (B,H,S,D)
    float* out_attn = out_ctx + (size_t)BB * HH * SS * DD;            // (B,H,S,S)

    const size_t smem =
        (size_t)(QTILE * SPAD + QTILE * DD + 128 + 16 + 16) * sizeof(float); // ~136 KB

    (void)hipFuncSetAttribute(reinterpret_cast<const void*>(attn_f32_wmma),
                              hipFuncAttributeMaxDynamicSharedMemorySize, (int)smem);

    dim3 grid(BB * HH * (SS / QTILE));  // 4096 blocks, q-tile fastest
    dim3 block(128);                    // 4 waves (wave32)
    attn_f32_wmma<<<grid, block, smem, stream>>>(Q, K, V, Mask, out_ctx, out_attn);
}